// MsgModelDiff_84189948936386
// MI455X (gfx1250) — compile-verified
//
#include <hip/hip_runtime.h>

typedef __attribute__((ext_vector_type(2))) float v2f;
typedef __attribute__((ext_vector_type(8))) float v8f;

#define NNODES 25000
#define CMSG 200
#define NT1 13            // ceil(200/16) tiles over message channels
#define NC1 (NT1*16)      // 208 padded message channels
#define NW  8             // waves per block (wave32 -> 256 threads)
#define NTHREADS (NW*32)

static __device__ __forceinline__ v8f wmma_f32(v2f a, v2f b, v8f c) {
    // D = A(16x4 f32) x B(4x16 f32) + C(16x16 f32)
    return __builtin_amdgcn_wmma_f32_16x16x4_f32(false, a, false, b, (short)0, c, false, false);
}

static __device__ __forceinline__ void atomAddF(float* p, float v) {
    unsafeAtomicAdd(p, v);   // hardware GLOBAL_ATOMIC_ADD_F32
}

// y[n,:] = (relu?)x[n,:] @ W1 + b1   (tiny: 20 MFLOP total, plain VALU)
template<int CIN, int COUT, bool RELU_IN>
__global__ __launch_bounds__(128) void mpd_node_kernel(
    const float* __restrict__ x, const float* __restrict__ W1,
    const float* __restrict__ b1, float* __restrict__ y, int N)
{
    const int total = N * COUT;
    for (int i = blockIdx.x * 128 + threadIdx.x; i < total; i += gridDim.x * 128) {
        const int n = i / COUT, j = i - n * COUT;
        float acc = b1[j];
        #pragma unroll
        for (int k = 0; k < CIN; ++k) {
            float v = x[n * CIN + k];
            if (RELU_IN) v = fmaxf(v, 0.f);
            acc = fmaf(v, W1[k * COUT + j], acc);
        }
        y[i] = acc;
    }
}

// Fused per-edge MLP + scatter-add. One wave per 16-edge tile, fp32 WMMA.
template<int CIN, int COUT, bool RELU_IN>
__global__ __launch_bounds__(NTHREADS) void mpd_edge_kernel(
    const float* __restrict__ x,
    const int* __restrict__ srcIdx, const int* __restrict__ dstIdx,
    const float* __restrict__ Wm1, const float* __restrict__ bm1,
    const float* __restrict__ Wm2, const float* __restrict__ bm2,
    const float* __restrict__ W2,  const float* __restrict__ b2,
    float* __restrict__ y, int E)
{
    constexpr int K1  = 2 * CIN;            // GEMM1 K
    constexpr int KP1 = (K1 + 3) & ~3;      // padded to x4
    constexpr int KC1 = KP1 / 4;            // K chunks
    constexpr int NP2 = (CIN + 15) & ~15;   // GEMM2 N padded
    constexpr int NT2 = NP2 / 16;
    constexpr int KP3 = (CIN + 3) & ~3;     // GEMM3 K padded
    constexpr int KC3 = KP3 / 4;
    constexpr int NO  = (COUT + 15) & ~15;  // GEMM3 N padded
    constexpr int NT3 = NO / 16;

    // Shared weights (padded, +1 col to break bank conflicts) + per-wave tiles.
    __shared__ float sWm1[KP1][NC1 + 1];    // GEMM1 B
    __shared__ float sWm2[NC1][NP2 + 1];    // GEMM2 B (rows >= 200 zero)
    __shared__ float sW2 [KP3][NO + 1];     // GEMM3 B
    __shared__ float sBm1[NC1];
    __shared__ float sBm2[NP2];
    __shared__ float sB2 [NO];
    __shared__ float sA[NW][16][KP1 + 1];   // per-wave [x_i | x_j] tile
    __shared__ float sT[NW][16][20];        // per-wave GEMM1->GEMM2 transpose bounce
    __shared__ float sG[NW][16][KP3 + 1];   // per-wave gate*(xi-xj) tile
    __shared__ int   sDst[NW][16];

    const int tid = threadIdx.x;
    for (int i = tid; i < KP1 * NC1; i += NTHREADS) {
        int r = i / NC1, c = i - r * NC1;
        sWm1[r][c] = (r < K1 && c < CMSG) ? Wm1[r * CMSG + c] : 0.f;
    }
    for (int i = tid; i < NC1 * NP2; i += NTHREADS) {
        int r = i / NP2, c = i - r * NP2;
        sWm2[r][c] = (r < CMSG && c < CIN) ? Wm2[r * CIN + c] : 0.f;
    }
    for (int i = tid; i < KP3 * NO; i += NTHREADS) {
        int r = i / NO, c = i - r * NO;
        sW2[r][c] = (r < CIN && c < COUT) ? W2[r * COUT + c] : 0.f;
    }
    for (int i = tid; i < NC1; i += NTHREADS) sBm1[i] = (i < CMSG) ? bm1[i] : 0.f;
    for (int i = tid; i < NP2; i += NTHREADS) sBm2[i] = (i < CIN)  ? bm2[i] : 0.f;
    for (int i = tid; i < NO;  i += NTHREADS) sB2[i]  = (i < COUT) ? b2[i]  : 0.f;
    __syncthreads();

    const int wave = tid >> 5;
    const int lane = tid & 31;        // wave32
    const int lm   = lane & 15;
    const int hi   = lane >> 4;       // 0: K-low half / x_i rows, 1: K-high half / x_j rows

    const int numTiles = (E + 15) >> 4;
    for (int tile = blockIdx.x * NW + wave; tile < numTiles; tile += gridDim.x * NW) {
        const int e  = tile * 16 + lm;
        const bool ev = (e < E);
        int node = 0;
        if (ev) node = hi ? srcIdx[e] : dstIdx[e];
        if (hi == 0) sDst[wave][lm] = ev ? node : -1;

        // Gather node row into sA: cols [0,CIN)=x_i, [CIN,2CIN)=x_j
        if constexpr (CIN % 4 == 0) {
            const float4* row = (const float4*)(x + (size_t)node * CIN);
            #pragma unroll
            for (int k4 = 0; k4 < CIN / 4; ++k4) {
                float4 v = row[k4];
                if (!ev) v = make_float4(0.f, 0.f, 0.f, 0.f);
                if (RELU_IN) {
                    v.x = fmaxf(v.x, 0.f); v.y = fmaxf(v.y, 0.f);
                    v.z = fmaxf(v.z, 0.f); v.w = fmaxf(v.w, 0.f);
                }
                sA[wave][lm][hi * CIN + 4 * k4 + 0] = v.x;
                sA[wave][lm][hi * CIN + 4 * k4 + 1] = v.y;
                sA[wave][lm][hi * CIN + 4 * k4 + 2] = v.z;
                sA[wave][lm][hi * CIN + 4 * k4 + 3] = v.w;
            }
        } else {
            #pragma unroll
            for (int k = 0; k < CIN; ++k) {
                float v = x[(size_t)node * CIN + k];
                if (!ev) v = 0.f;
                if (RELU_IN) v = fmaxf(v, 0.f);
                sA[wave][lm][hi * CIN + k] = v;
            }
        }
        if (hi == 0) {
            #pragma unroll
            for (int k = K1; k < KP1; ++k) sA[wave][lm][k] = 0.f;  // K padding
        }

        // ---- GEMM1 (16xK1 @ K1x208) fused with GEMM2 (16x208 @ 208xNP2) ----
        v8f accG[NT2];
        #pragma unroll
        for (int t = 0; t < NT2; ++t) accG[t] = {};

        for (int t1 = 0; t1 < NT1; ++t1) {
            v8f acc = {};
            #pragma unroll
            for (int c = 0; c < KC1; ++c) {
                const int ka = 4 * c + 2 * hi;
                v2f a, b;
                a.x = sA[wave][lm][ka];
                a.y = sA[wave][lm][ka + 1];
                b.x = sWm1[ka][t1 * 16 + lm];
                b.y = sWm1[ka + 1][t1 * 16 + lm];
                acc = wmma_f32(a, b, acc);
            }
            // bias + relu, C-layout -> sT (per-wave transpose bounce)
            const float bb = sBm1[t1 * 16 + lm];
            #pragma unroll
            for (int r = 0; r < 8; ++r) {
                sT[wave][r + 8 * hi][lm] = fmaxf(acc[r] + bb, 0.f);
            }
            // GEMM2 partial over this 16-wide K block
            #pragma unroll
            for (int cc = 0; cc < 4; ++cc) {
                const int kk = 4 * cc + 2 * hi;
                const int kg = t1 * 16 + kk;
                v2f a2;
                a2.x = sT[wave][lm][kk];
                a2.y = sT[wave][lm][kk + 1];
                #pragma unroll
                for (int t2 = 0; t2 < NT2; ++t2) {
                    v2f b2v;
                    b2v.x = sWm2[kg][t2 * 16 + lm];
                    b2v.y = sWm2[kg + 1][t2 * 16 + lm];
                    accG[t2] = wmma_f32(a2, b2v, accG[t2]);
                }
            }
        }

        // ---- gate = accG + bm2 ; sG = gate * (x_i - x_j), zero-padded K ----
        #pragma unroll
        for (int t2 = 0; t2 < NT2; ++t2) {
            const int col = t2 * 16 + lm;
            const float bv = sBm2[col];
            #pragma unroll
            for (int r = 0; r < 8; ++r) {
                const int M = r + 8 * hi;
                const float g = accG[t2][r] + bv;
                const float dv = (col < CIN)
                    ? (sA[wave][M][col] - sA[wave][M][CIN + col]) : 0.f;
                if (col < KP3) sG[wave][M][col] = g * dv;
            }
        }

        // ---- GEMM3 (16xKP3 @ KP3xNO) + bias + atomic scatter-add by dst ----
        #pragma unroll
        for (int t3 = 0; t3 < NT3; ++t3) {
            v8f acc3 = {};
            #pragma unroll
            for (int c = 0; c < KC3; ++c) {
                const int ka = 4 * c + 2 * hi;
                v2f a3, b3;
                a3.x = sG[wave][lm][ka];
                a3.y = sG[wave][lm][ka + 1];
                b3.x = sW2[ka][t3 * 16 + lm];
                b3.y = sW2[ka + 1][t3 * 16 + lm];
                acc3 = wmma_f32(a3, b3, acc3);
            }
            const int col = t3 * 16 + lm;
            if (col < COUT) {
                const float bb = sB2[col];
                #pragma unroll
                for (int r = 0; r < 8; ++r) {
                    const int M = r + 8 * hi;
                    const int d = sDst[wave][M];
                    if (d >= 0) atomAddF(&y[(size_t)d * COUT + col], acc3[r] + bb);
                }
            }
        }
    }
}

extern "C" void kernel_launch(void* const* d_in, const int* in_sizes, int n_in,
                              void* d_out, int out_size, void* d_ws, size_t ws_size,
                              hipStream_t stream)
{
    (void)n_in; (void)out_size; (void)ws_size;
    const float* features = (const float*)d_in[0];
    const int*   edges    = (const int*)d_in[1];   // [2, E] (weights d_in[2] unused by reference)
    const float* P[24];
    for (int i = 0; i < 24; ++i) P[i] = (const float*)d_in[3 + i];
    // P[0..7]=d_{W1,b1,Wm1,bm1,Wm2,bm2,W2,b2}, P[8..15]=h_*, P[16..23]=o_*

    const int E = in_sizes[1] / 2;
    const int N = NNODES;
    const int* srcI = edges;        // edges[0] = src
    const int* dstI = edges + E;    // edges[1] = dst

    float* y0  = (float*)d_ws;
    float* y1  = y0 + (size_t)N * 20;
    float* out = (float*)d_out;

    const int tiles   = (E + 15) / 16;
    const int eblocks = (tiles + NW - 1) / NW;
    auto nb = [](int total) { int b = (total + 127) / 128; return b > 4096 ? 4096 : b; };

    // layer d: 1 -> 20 (no input relu)
    mpd_node_kernel<1, 20, false><<<nb(N * 20), 128, 0, stream>>>(features, P[0], P[1], y0, N);
    mpd_edge_kernel<1, 20, false><<<eblocks, NTHREADS, 0, stream>>>(
        features, srcI, dstI, P[2], P[3], P[4], P[5], P[6], P[7], y0, E);

    // 3 hidden layers: 20 -> 20 (relu folded into gathers)
    mpd_node_kernel<20, 20, true><<<nb(N * 20), 128, 0, stream>>>(y0, P[8], P[9], y1, N);
    mpd_edge_kernel<20, 20, true><<<eblocks, NTHREADS, 0, stream>>>(
        y0, srcI, dstI, P[10], P[11], P[12], P[13], P[14], P[15], y1, E);

    mpd_node_kernel<20, 20, true><<<nb(N * 20), 128, 0, stream>>>(y1, P[8], P[9], y0, N);
    mpd_edge_kernel<20, 20, true><<<eblocks, NTHREADS, 0, stream>>>(
        y1, srcI, dstI, P[10], P[11], P[12], P[13], P[14], P[15], y0, E);

    mpd_node_kernel<20, 20, true><<<nb(N * 20), 128, 0, stream>>>(y0, P[8], P[9], y1, N);
    mpd_edge_kernel<20, 20, true><<<eblocks, NTHREADS, 0, stream>>>(
        y0, srcI, dstI, P[10], P[11], P[12], P[13], P[14], P[15], y1, E);

    // output layer o: 20 -> 1 (input relu, no output relu)
    mpd_node_kernel<20, 1, true><<<nb(N), 128, 0, stream>>>(y1, P[16], P[17], out, N);
    mpd_edge_kernel<20, 1, true><<<eblocks, NTHREADS, 0, stream>>>(
        y1, srcI, dstI, P[18], P[19], P[20], P[21], P[22], P[23], out, E);
}